// KBAdapter_6648609374618
// MI455X (gfx1250) — compile-verified
//
#include <hip/hip_runtime.h>
#include <math.h>

// ---------------------------------------------------------------------------
// KB adapter on MI455X (gfx1250, wave32, WMMA + Tensor Data Mover).
// Sizes: M=B*S=1024, H=2048, D=512, N=131072.
// Output: query[1024*512] | probs[1024*131072] | context[1024*512].
// Pipeline (fp32, V_WMMA_F32_16X16X4_F32; TDM double-buffered LDS staging):
//   k_init    : zero context region, init per-row max (monotone-uint) in ws
//   k_qln     : query projection + bias + LayerNorm
//   k_logits  : WMMA GEMM q @ K^T (TDM-staged tiles) -> logits + row max
//   k_expsum  : probs = exp(logit - max), per-row 1/sum into ws
//   k_ctx     : WMMA GEMM exp-probs @ V (TDM-staged tiles) -> atomic f32 adds
//   k_scale   : probs *= 1/sum ; context *= 1/sum
// ---------------------------------------------------------------------------

typedef __attribute__((ext_vector_type(2))) float v2f;
typedef __attribute__((ext_vector_type(8))) float v8f;
typedef unsigned int u32;
typedef __attribute__((ext_vector_type(4))) u32 v4u;
typedef __attribute__((ext_vector_type(8))) int v8i;
typedef __attribute__((ext_vector_type(4))) int v4i;

#define M_ROWS 1024
#define HID    2048
#define NKB    131072
#define DD     512
#define LN_EPS 1e-5f

// Monotonic float<->uint mapping so we can use integer atomic max on floats.
__device__ __forceinline__ unsigned f2mono(float f) {
  unsigned u = __float_as_uint(f);
  return (u & 0x80000000u) ? ~u : (u | 0x80000000u);
}
__device__ __forceinline__ float mono2f(unsigned u) {
  unsigned b = (u & 0x80000000u) ? (u ^ 0x80000000u) : ~u;
  return __uint_as_float(b);
}

// Low 32 bits of a generic pointer to __shared__ = LDS byte offset
// (LDS aperture places the as3 offset in addr[31:0]).
__device__ __forceinline__ u32 lds_off(const void* p) {
  return (u32)(unsigned long long)(size_t)p;
}

// ---------------------------------------------------------------------------
// TDM: issue a 2D-tile tensor_load_to_lds. Row-major source tensor of 4-byte
// elements; tile = tile_d1 rows x tile_d0 contiguous elements, row stride
// stride0 elements. Tensor dims are set to the tile dims (tiles are always
// fully in-bounds), groups 2/3 are zero (2D). Tracked by TENSORcnt.
// D# bitfields per CDNA5 ISA ch.8 (group0: count/lds_addr/global_addr/type,
// group1: data_size/dims/tile dims/strides).
// ---------------------------------------------------------------------------
__device__ __forceinline__ void tdm_load_2d(const void* gptr, u32 lds_addr,
                                            u32 tile_d0, u32 tile_d1,
                                            unsigned long long stride0_elems) {
  unsigned long long ga = (unsigned long long)(size_t)gptr;
  v4u g0;
  g0[0] = 1u;                                            // count=1, user D#
  g0[1] = lds_addr;                                      // lds_addr (bytes)
  g0[2] = (u32)ga;                                       // global_addr[31:0]
  g0[3] = (u32)((ga >> 32) & 0x01ffffffu) | (2u << 30);  // [56:32] + type=2
  v8i g1;
  g1[0] = (int)(2u << 16);                               // data_size=2 (4B)
  g1[1] = (int)((tile_d0 & 0xffffu) << 16);              // tensor_dim0 lo16
  g1[2] = (int)((tile_d0 >> 16) | ((tile_d1 & 0xffffu) << 16)); // d0 hi|d1 lo
  g1[3] = (int)((tile_d1 >> 16) | (tile_d0 << 16));      // d1 hi | tile_dim0
  g1[4] = (int)(tile_d1 & 0xffffu);                      // tile_dim1 (dim2=0)
  g1[5] = (int)(u32)stride0_elems;                       // dim0_stride[31:0]
  g1[6] = (int)(u32)((stride0_elems >> 32) & 0xffffu);   // dim0_stride[47:32]
  g1[7] = 0;                                             // dim1_stride unused
  v4i zz = {0, 0, 0, 0};
#if defined(__clang_major__) && (__clang_major__ >= 23)
  v8i z8 = {0, 0, 0, 0, 0, 0, 0, 0};
  __builtin_amdgcn_tensor_load_to_lds(g0, g1, zz, zz, z8, 0);
#else
  __builtin_amdgcn_tensor_load_to_lds(g0, g1, zz, zz, 0);
#endif
}

// ---------------------------------------------------------------------------
__global__ void __launch_bounds__(256) k_init(float* __restrict__ ctx,
                                              unsigned* __restrict__ wsmax,
                                              int nctx) {
  int gid = blockIdx.x * blockDim.x + threadIdx.x;
  float4* c4 = (float4*)ctx;
  int n4 = nctx >> 2;
  for (int i = gid; i < n4; i += gridDim.x * blockDim.x)
    c4[i] = make_float4(0.f, 0.f, 0.f, 0.f);
  if (gid < M_ROWS) wsmax[gid] = 0u;  // smallest monotone code
}

// ---------------------------------------------------------------------------
// Query projection + LayerNorm. One block = 16 rows; hs tile staged in LDS in
// 512-wide k-chunks; each thread owns 2 output columns for all 16 rows.
__global__ void __launch_bounds__(256) k_qln(const float* __restrict__ hs,
                                             const float* __restrict__ Wq,
                                             const float* __restrict__ bq,
                                             float* __restrict__ qout) {
  __shared__ float smem[16 * 512];  // 32 KB, reused for LN reductions
  const int tid = threadIdx.x;
  const int m0 = blockIdx.x * 16;
  const int d0 = tid, d1 = tid + 256;

  float acc0[16], acc1[16];
#pragma unroll
  for (int r = 0; r < 16; ++r) { acc0[r] = 0.f; acc1[r] = 0.f; }

  const float4* h4 = (const float4*)hs;
  const float4* w4 = (const float4*)Wq;
  for (int kc = 0; kc < 4; ++kc) {
    __syncthreads();
#pragma unroll
    for (int t = 0; t < 8; ++t) {
      int idx = tid + t * 256;           // 0..2047 float4 of 16x512 tile
      int row = idx >> 7, c = idx & 127;
      ((float4*)smem)[row * 128 + c] = h4[(size_t)(m0 + row) * 512 + kc * 128 + c];
    }
    __syncthreads();
    for (int h = 0; h < 128; ++h) {
      float4 w0 = w4[(size_t)d0 * 512 + kc * 128 + h];
      float4 w1 = w4[(size_t)d1 * 512 + kc * 128 + h];
#pragma unroll
      for (int r = 0; r < 16; ++r) {
        float4 hv = ((const float4*)smem)[r * 128 + h];
        acc0[r] += hv.x * w0.x + hv.y * w0.y + hv.z * w0.z + hv.w * w0.w;
        acc1[r] += hv.x * w1.x + hv.y * w1.y + hv.z * w1.z + hv.w * w1.w;
      }
    }
  }
  float b0 = bq[d0], b1 = bq[d1];
#pragma unroll
  for (int r = 0; r < 16; ++r) { acc0[r] += b0; acc1[r] += b1; }

  __syncthreads();
#pragma unroll
  for (int r = 0; r < 16; ++r) {
    smem[r * 256 + tid]        = acc0[r] + acc1[r];
    smem[4096 + r * 256 + tid] = acc0[r] * acc0[r] + acc1[r] * acc1[r];
  }
  __syncthreads();
  for (int off = 128; off > 0; off >>= 1) {
    if (tid < off) {
#pragma unroll
      for (int r = 0; r < 16; ++r) {
        smem[r * 256 + tid]        += smem[r * 256 + tid + off];
        smem[4096 + r * 256 + tid] += smem[4096 + r * 256 + tid + off];
      }
    }
    __syncthreads();
  }
#pragma unroll
  for (int r = 0; r < 16; ++r) {
    float mean = smem[r * 256] * (1.0f / 512.0f);
    float var  = smem[4096 + r * 256] * (1.0f / 512.0f) - mean * mean;
    float rstd = rsqrtf(var + LN_EPS);
    qout[(size_t)(m0 + r) * DD + d0] = (acc0[r] - mean) * rstd;
    qout[(size_t)(m0 + r) * DD + d1] = (acc1[r] - mean) * rstd;
  }
}

// ---------------------------------------------------------------------------
// Logits GEMM: workgroup = 128 rows x 1024 KB-columns, iteration = one
// (round, kc): round = 128-col group (one 16-col tile per wave), kc = 64-wide
// k chunk. Q(128x64) and K(128x64) tiles are TDM-loaded into double-buffered
// LDS by wave 0 and overlapped with WMMA compute; per-row max folded into the
// epilogue (lane shuffles -> LDS atomic max -> global atomic max).
__global__ void __launch_bounds__(256) k_logits(const float* __restrict__ q,
                                                const float* __restrict__ kb,
                                                float* __restrict__ probs,
                                                unsigned* __restrict__ wsmax) {
  __shared__ float sQ[2][128 * 64];   // 2 x 32 KB
  __shared__ float sK[2][128 * 64];   // 2 x 32 KB
  __shared__ unsigned smax[128];
  const int tid  = threadIdx.x;
  const int lane = tid & 31, wave = tid >> 5;
  const int half = lane >> 4, l16 = lane & 15;
  const int rowbase = blockIdx.x * 128;
  const int nsplit  = blockIdx.y;  // 0..127 -> 1024 columns each

  if (tid < 128) smax[tid] = 0u;

  auto issue = [&](int it, int buf) {
    const int round = it >> 3, kc = it & 7;
    tdm_load_2d(q + (size_t)rowbase * DD + kc * 64,
                lds_off(&sQ[buf][0]), 64u, 128u, (unsigned long long)DD);
    tdm_load_2d(kb + (size_t)(nsplit * 1024 + round * 128) * DD + kc * 64,
                lds_off(&sK[buf][0]), 64u, 128u, (unsigned long long)DD);
  };

  if (wave == 0) issue(0, 0);

  const v8f zacc = {};
  v8f acc[8];
  for (int it = 0; it < 64; ++it) {
    const int buf = it & 1;
    const int round = it >> 3, kc = it & 7;
    if (kc == 0) {
#pragma unroll
      for (int rt = 0; rt < 8; ++rt) acc[rt] = zacc;
    }
    if (wave == 0) __builtin_amdgcn_s_wait_tensorcnt(0);
    __syncthreads();
    if (wave == 0 && it + 1 < 64) issue(it + 1, buf ^ 1);

#pragma unroll
    for (int kk = 0; kk < 64; kk += 4) {
      v2f b = *(const v2f*)&sK[buf][(wave * 16 + l16) * 64 + kk + 2 * half];
#pragma unroll
      for (int rt = 0; rt < 8; ++rt) {
        v2f a = *(const v2f*)&sQ[buf][(rt * 16 + l16) * 64 + kk + 2 * half];
        acc[rt] = __builtin_amdgcn_wmma_f32_16x16x4_f32(
            false, a, false, b, (short)0, acc[rt], false, false);
      }
    }

    if (kc == 7) {  // round complete: store logits + fold row max
      const int ncol = nsplit * 1024 + round * 128 + wave * 16 + l16;
#pragma unroll
      for (int rt = 0; rt < 8; ++rt) {
#pragma unroll
        for (int r = 0; r < 8; ++r) {
          int m = rowbase + rt * 16 + r + 8 * half;
          float v = acc[rt][r];
          probs[(size_t)m * NKB + ncol] = v;
          float mx = v;
          mx = fmaxf(mx, __shfl_xor(mx, 1, 32));
          mx = fmaxf(mx, __shfl_xor(mx, 2, 32));
          mx = fmaxf(mx, __shfl_xor(mx, 4, 32));
          mx = fmaxf(mx, __shfl_xor(mx, 8, 32));
          if (l16 == 0) atomicMax(&smax[rt * 16 + r + 8 * half], f2mono(mx));
        }
      }
    }
  }
  __syncthreads();
  if (tid < 128) atomicMax(&wsmax[rowbase + tid], smax[tid]);
}

// ---------------------------------------------------------------------------
// probs = exp(logit - rowmax); one block per row; 1/sum -> ws.
__global__ void __launch_bounds__(256) k_expsum(float* __restrict__ probs,
                                                const unsigned* __restrict__ wsmax,
                                                float* __restrict__ wsinv) {
  __shared__ float sred[256];
  const int m = blockIdx.x, tid = threadIdx.x;
  const float mx = mono2f(wsmax[m]);
  float4* p4 = (float4*)(probs + (size_t)m * NKB);
  float s = 0.f;
  for (int i = tid; i < NKB / 4; i += 256) {
    float4 v = p4[i];
    v.x = expf(v.x - mx); v.y = expf(v.y - mx);
    v.z = expf(v.z - mx); v.w = expf(v.w - mx);
    s += v.x + v.y + v.z + v.w;
    p4[i] = v;
  }
  sred[tid] = s;
  __syncthreads();
  for (int off = 128; off > 0; off >>= 1) {
    if (tid < off) sred[tid] += sred[tid + off];
    __syncthreads();
  }
  if (tid == 0) wsinv[m] = 1.0f / sred[0];
}

// ---------------------------------------------------------------------------
// Context GEMM: workgroup = 128 rows x 256 d-cols x 16384-n chunk.
// P(128x32) + V(32x256) tiles TDM-loaded into double-buffered LDS by wave 0,
// overlapped with compute; each wave owns 2 d-tiles x 8 row-tiles = 16 WMMA
// accumulators; cross-n-split combine via native f32 atomic adds.
__global__ void __launch_bounds__(256) k_ctx(const float* __restrict__ probs,
                                             const float* __restrict__ vb,
                                             float* __restrict__ ctx) {
  __shared__ float sP[2][128 * 32];   // 2 x 16 KB
  __shared__ float sV[2][32 * 256];   // 2 x 32 KB
  const int tid  = threadIdx.x;
  const int lane = tid & 31, wave = tid >> 5;
  const int half = lane >> 4, l16 = lane & 15;
  const int rowbase = blockIdx.x * 128;
  const int dbase   = blockIdx.y * 256;
  const int nbase0  = blockIdx.z * 16384;

  auto issue = [&](int it, int buf) {
    const int nb = nbase0 + it * 32;
    tdm_load_2d(probs + (size_t)rowbase * NKB + nb,
                lds_off(&sP[buf][0]), 32u, 128u, (unsigned long long)NKB);
    tdm_load_2d(vb + (size_t)nb * DD + dbase,
                lds_off(&sV[buf][0]), 256u, 32u, (unsigned long long)DD);
  };

  if (wave == 0) issue(0, 0);

  v8f acc[16] = {};
  for (int it = 0; it < 512; ++it) {
    const int buf = it & 1;
    if (wave == 0) __builtin_amdgcn_s_wait_tensorcnt(0);
    __syncthreads();
    if (wave == 0 && it + 1 < 512) issue(it + 1, buf ^ 1);

#pragma unroll
    for (int nn = 0; nn < 32; nn += 4) {
      v2f a[8];
#pragma unroll
      for (int rt = 0; rt < 8; ++rt)
        a[rt] = *(const v2f*)&sP[buf][(rt * 16 + l16) * 32 + nn + 2 * half];
#pragma unroll
      for (int j = 0; j < 2; ++j) {
        const int dcol = (wave * 2 + j) * 16 + l16;
        v2f b;
        b.x = sV[buf][(nn + 2 * half) * 256 + dcol];
        b.y = sV[buf][(nn + 2 * half + 1) * 256 + dcol];
#pragma unroll
        for (int rt = 0; rt < 8; ++rt) {
          acc[j * 8 + rt] = __builtin_amdgcn_wmma_f32_16x16x4_f32(
              false, a[rt], false, b, (short)0, acc[j * 8 + rt], false, false);
        }
      }
    }
  }
#pragma unroll
  for (int j = 0; j < 2; ++j) {
    const int d = dbase + (wave * 2 + j) * 16 + l16;
#pragma unroll
    for (int rt = 0; rt < 8; ++rt) {
#pragma unroll
      for (int r = 0; r < 8; ++r) {
        int m = rowbase + rt * 16 + r + 8 * half;
        atomicAdd(&ctx[(size_t)m * DD + d], acc[j * 8 + rt][r]);
      }
    }
  }
}

// ---------------------------------------------------------------------------
__global__ void __launch_bounds__(256) k_scale(float* __restrict__ probs,
                                               float* __restrict__ ctx,
                                               const float* __restrict__ wsinv) {
  const int m = blockIdx.x, tid = threadIdx.x;
  const float inv = wsinv[m];
  float4* p4 = (float4*)(probs + (size_t)m * NKB);
  for (int i = tid; i < NKB / 4; i += 256) {
    float4 v = p4[i];
    v.x *= inv; v.y *= inv; v.z *= inv; v.w *= inv;
    p4[i] = v;
  }
  ctx[(size_t)m * DD + tid] *= inv;
  ctx[(size_t)m * DD + tid + 256] *= inv;
}

// ---------------------------------------------------------------------------
extern "C" void kernel_launch(void* const* d_in, const int* in_sizes, int n_in,
                              void* d_out, int out_size, void* d_ws, size_t ws_size,
                              hipStream_t stream) {
  (void)in_sizes; (void)n_in; (void)out_size; (void)ws_size;
  const float* hs = (const float*)d_in[0];  // [1024, 2048]
  const float* kb = (const float*)d_in[1];  // [131072, 512]
  const float* vb = (const float*)d_in[2];  // [131072, 512]
  const float* Wq = (const float*)d_in[3];  // [512, 2048]
  const float* bq = (const float*)d_in[4];  // [512]

  float* out   = (float*)d_out;
  float* qout  = out;                                   // [1024, 512]
  float* probs = out + (size_t)M_ROWS * DD;             // [1024, 131072]
  float* ctx   = probs + (size_t)M_ROWS * NKB;          // [1024, 512]

  unsigned* wsmax = (unsigned*)d_ws;          // [1024] monotone row max
  float*    wsinv = (float*)d_ws + M_ROWS;    // [1024] 1/sum

  k_init  <<<512, 256, 0, stream>>>(ctx, wsmax, M_ROWS * DD);
  k_qln   <<<64, 256, 0, stream>>>(hs, Wq, bq, qout);
  k_logits<<<dim3(8, 128), 256, 0, stream>>>(qout, kb, probs, wsmax);
  k_expsum<<<1024, 256, 0, stream>>>(probs, wsmax, wsinv);
  k_ctx   <<<dim3(8, 2, 8), 256, 0, stream>>>(probs, vb, ctx);
  k_scale <<<1024, 256, 0, stream>>>(probs, ctx, wsinv);
}